// MoE_16664473108485
// MI455X (gfx1250) — compile-verified
//
#include <hip/hip_runtime.h>
#include <cmath>
#include <stdint.h>

typedef __attribute__((ext_vector_type(2))) float v2f;
typedef __attribute__((ext_vector_type(4))) float v4f;
typedef __attribute__((ext_vector_type(8))) float v8f;

#define EPSV 1e-5f
#define TB 32           // samples per workgroup tile
#define AS_STRIDE 1028  // 1024 + 4 pad -> conflict-free WMMA A fragment b64 loads
#define HB_STRIDE 132   // 128 + 4 pad

__global__ __launch_bounds__(256)
void moe_gate_fused(const float* __restrict__ features,   // [E=8, B, D=128]
                    const float* __restrict__ w1,         // [1024, 128]
                    const float* __restrict__ b1,         // [128]
                    const float* __restrict__ ln1g,       // [128]
                    const float* __restrict__ ln1b,       // [128]
                    const float* __restrict__ w2,         // [128, 8]
                    const float* __restrict__ b2,         // [8]
                    const float* __restrict__ outg,       // [128]
                    const float* __restrict__ outb,       // [128]
                    float* __restrict__ out,              // [E, B, D]
                    int Bn)
{
    extern __shared__ float smem[];
    float* As = smem;                        // TB x AS_STRIDE  (feature tile, row-major)
    float* Hb = As + TB * AS_STRIDE;         // TB x HB_STRIDE  (GELU(h))
    float* Sc = Hb + TB * HB_STRIDE;         // TB x 8          (softmax scores)

    const int t    = threadIdx.x;
    const int w    = t >> 5;                 // wave id 0..7
    const int lane = t & 31;
    const int b0   = blockIdx.x * TB;

    // ---------------- Phase A: features tile -> LDS via async copy (ASYNCcnt) ------
    // iteration i: all 256 lanes issue one async b128 for row i; wave w covers
    // expert w's contiguous 512B, so each request is fully coalesced.
    {
        const int c4 = t << 2;               // column 0..1023, step 4
        const int e  = c4 >> 7;
        const int d  = c4 & 127;
        const unsigned lds0 = (unsigned)(uintptr_t)(&As[c4]);         // row 0 slot
        const float*   g0   = features + (((size_t)e * Bn + b0) << 7) + d;
        for (int i = 0; i < TB; ++i) {
            const unsigned     ldso = lds0 + (unsigned)(i * (AS_STRIDE * 4));
            const float* const gp   = g0 + ((size_t)i << 7);
            asm volatile("global_load_async_to_lds_b128 %0, %1, off"
                         :: "v"(ldso), "v"(gp) : "memory");
        }
        asm volatile("s_wait_asynccnt 0x0" ::: "memory");
    }
    __syncthreads();

    // ---------------- Phase B: h_pre = A @ W1 via V_WMMA_F32_16X16X4_F32 -----------
    const int n0    = w << 4;                // this wave's 16-column N tile of H=128
    const int nn    = n0 + (lane & 15);      // C/D column owned by this lane
    const int khalf = (lane >> 4) << 1;      // 0 or 2 (A/B fragment K split)

    v8f acc0 = {0.f,0.f,0.f,0.f,0.f,0.f,0.f,0.f};  // rows 0..15
    v8f acc1 = {0.f,0.f,0.f,0.f,0.f,0.f,0.f,0.f};  // rows 16..31

    const float* wp     = w1 + nn;                              // column nn of W1
    const float* a_base = &As[(lane & 15) * AS_STRIDE + khalf]; // 8B aligned

    #pragma unroll 4
    for (int kb = 0; kb < 256; ++kb) {
        const int k0 = (kb << 2) + khalf;
        v2f bf;                                   // B frag: B[k0..k0+1][nn]
        bf.x = wp[(size_t)k0 << 7];
        bf.y = wp[(size_t)(k0 + 1) << 7];
        v2f a0 = *(const v2f*)(a_base + (kb << 2));                    // rows 0..15
        v2f a1 = *(const v2f*)(a_base + (kb << 2) + 16 * AS_STRIDE);   // rows 16..31
        acc0 = __builtin_amdgcn_wmma_f32_16x16x4_f32(false, a0, false, bf,
                                                     (short)0, acc0, false, false);
        acc1 = __builtin_amdgcn_wmma_f32_16x16x4_f32(false, a1, false, bf,
                                                     (short)0, acc1, false, false);
    }

    // ---------------- Phase C: bias + exact GELU -> LDS ----------------------------
    {
        const float b1n   = b1[nn];
        const int   rbase = (lane >> 4) << 3;    // C layout: lanes 16..31 hold M=8..15
        #pragma unroll
        for (int v = 0; v < 8; ++v) {
            float x0 = acc0[v] + b1n;
            Hb[(rbase + v) * HB_STRIDE + nn] =
                0.5f * x0 * (1.0f + erff(x0 * 0.70710678118654752f));
            float x1 = acc1[v] + b1n;
            Hb[(16 + rbase + v) * HB_STRIDE + nn] =
                0.5f * x1 * (1.0f + erff(x1 * 0.70710678118654752f));
        }
    }
    __syncthreads();

    // ---------------- Phase D: LayerNorm(H) -> logits -> softmax -------------------
    // 8 lanes per row (row = t>>3, sub = t&7), each lane owns 16 of 128 columns.
    {
        const int row = t >> 3;
        const int sub = t & 7;
        float v[16];
        float sum = 0.f, sq = 0.f;
        const float* hr = &Hb[row * HB_STRIDE + (sub << 4)];
        #pragma unroll
        for (int i = 0; i < 16; ++i) { float x = hr[i]; v[i] = x; sum += x; sq += x * x; }
        #pragma unroll
        for (int m = 1; m <= 4; m <<= 1) {
            sum += __shfl_xor(sum, m, 32);
            sq  += __shfl_xor(sq,  m, 32);
        }
        const float mu  = sum * (1.0f / 128.0f);
        const float var = sq * (1.0f / 128.0f) - mu * mu;
        const float inv = rsqrtf(var + EPSV);
        #pragma unroll
        for (int i = 0; i < 16; ++i) {
            const int c = (sub << 4) + i;
            v[i] = (v[i] - mu) * inv * ln1g[c] + ln1b[c];
        }
        // logits: lane `sub` ends up holding logit[sub]
        float logit = 0.f;
        #pragma unroll
        for (int e = 0; e < 8; ++e) {
            float p = 0.f;
            #pragma unroll
            for (int i = 0; i < 16; ++i)
                p += v[i] * w2[((((sub << 4) + i)) << 3) + e];
            p += __shfl_xor(p, 1, 32);
            p += __shfl_xor(p, 2, 32);
            p += __shfl_xor(p, 4, 32);
            if (e == sub) logit = p + b2[e];
        }
        // softmax across the 8 lanes of this row
        float mx = logit;
        mx = fmaxf(mx, __shfl_xor(mx, 1, 32));
        mx = fmaxf(mx, __shfl_xor(mx, 2, 32));
        mx = fmaxf(mx, __shfl_xor(mx, 4, 32));
        float ex = __expf(logit - mx);
        float se = ex;
        se += __shfl_xor(se, 1, 32);
        se += __shfl_xor(se, 2, 32);
        se += __shfl_xor(se, 4, 32);
        Sc[(row << 3) + sub] = ex / se;
    }
    __syncthreads();

    // ---------------- Phase E: out = LN(score * x), streamed with NT stores --------
    // wave w handles expert e = w, all 32 rows; 32 lanes x v4f cover D=128.
    {
        const int e  = w;
        const int d0 = lane << 2;
        const v4f og = *(const v4f*)(outg + d0);
        const v4f ob = *(const v4f*)(outb + d0);
        for (int row = 0; row < TB; ++row) {
            const v4f x = *(const v4f*)(&As[row * AS_STRIDE + (e << 7) + d0]);
            float sum = x.x + x.y + x.z + x.w;
            float sq  = fmaf(x.x, x.x, fmaf(x.y, x.y, fmaf(x.z, x.z, x.w * x.w)));
            #pragma unroll
            for (int m = 1; m <= 16; m <<= 1) {
                sum += __shfl_xor(sum, m, 32);
                sq  += __shfl_xor(sq,  m, 32);
            }
            const float mu  = sum * (1.0f / 128.0f);
            const float var = sq * (1.0f / 128.0f) - mu * mu;
            const float s   = Sc[(row << 3) + e];
            const float a   = s * rsqrtf(s * s * var + EPSV);
            v4f o;
            o.x = (x.x - mu) * a * og.x + ob.x;
            o.y = (x.y - mu) * a * og.y + ob.y;
            o.z = (x.z - mu) * a * og.z + ob.z;
            o.w = (x.w - mu) * a * og.w + ob.w;
            __builtin_nontemporal_store(
                o, (v4f*)(out + (((size_t)e * Bn + b0 + row) << 7) + d0));
        }
    }
}

extern "C" void kernel_launch(void* const* d_in, const int* in_sizes, int n_in,
                              void* d_out, int out_size, void* d_ws, size_t ws_size,
                              hipStream_t stream) {
    const float* features = (const float*)d_in[0];
    const float* w1       = (const float*)d_in[1];
    const float* b1       = (const float*)d_in[2];
    const float* ln1g     = (const float*)d_in[3];
    const float* ln1b     = (const float*)d_in[4];
    const float* w2       = (const float*)d_in[5];
    const float* b2       = (const float*)d_in[6];
    const float* outg     = (const float*)d_in[7];
    const float* outb     = (const float*)d_in[8];
    float*       out      = (float*)d_out;

    const int Bn     = in_sizes[0] / (8 * 128);   // B = 131072
    const int blocks = Bn / TB;                   // 4096
    const size_t smem_bytes =
        (size_t)(TB * AS_STRIDE + TB * HB_STRIDE + TB * 8) * sizeof(float); // ~149.5 KB

    (void)hipFuncSetAttribute(reinterpret_cast<const void*>(moe_gate_fused),
                              hipFuncAttributeMaxDynamicSharedMemorySize,
                              (int)smem_bytes);

    moe_gate_fused<<<blocks, 256, smem_bytes, stream>>>(
        features, w1, b1, ln1g, ln1b, w2, b2, outg, outb, out, Bn);
}